// RGCNBlockLayer_10402410791332
// MI455X (gfx1250) — compile-verified
//
#include <hip/hip_runtime.h>

#define IN_F  20
#define OUT_F 20

typedef __attribute__((ext_vector_type(2))) float v2f;
typedef __attribute__((ext_vector_type(8))) float v8f;

// ---------------------------------------------------------------------------
// Kernel 1: zero the aggregation buffer (we accumulate straight into d_out).
// n is guaranteed divisible by 4 (n_nodes * 20), so vectorize with float4.
// ---------------------------------------------------------------------------
__global__ __launch_bounds__(256) void zero_out_kernel(float4* __restrict__ out, int n4) {
    int i = blockIdx.x * blockDim.x + threadIdx.x;
    int stride = gridDim.x * blockDim.x;
    float4 z = make_float4(0.f, 0.f, 0.f, 0.f);
    for (; i < n4; i += stride) out[i] = z;
}

// ---------------------------------------------------------------------------
// Kernel 2: per-edge message + gate + scatter-add (FP32 hardware atomics)
//   msg[b,o]  = sum_i h[src][b*5+i] * W[etype][b*25+i*5+o]
//   gate      = sigmoid(dot(h[src], gate_w[etype]) + gate_b[etype])
//   agg[dst] += gate * (msg + bias_term[etype])
// ---------------------------------------------------------------------------
__global__ __launch_bounds__(256) void edge_kernel(
    const float* __restrict__ h,          // [n_nodes][20]
    const float* __restrict__ weight,     // [200][100]  (4 bases x 5x5)
    const float* __restrict__ bias_term,  // [200][20]
    const float* __restrict__ gate_w,     // [200][20]   (gate_weight[...,0])
    const float* __restrict__ gate_b,     // [200]
    const int*   __restrict__ esrc,
    const int*   __restrict__ edst,
    const int*   __restrict__ etype,
    float*       __restrict__ agg,        // [n_nodes][20], pre-zeroed
    int n_edges)
{
    int e = blockIdx.x * blockDim.x + threadIdx.x;
    if (e >= n_edges) return;

    int s = esrc[e];
    int d = edst[e];
    int r = etype[e];

    const float* W   = weight    + (size_t)r * 100;
    const float* gwr = gate_w    + (size_t)r * OUT_F;
    const float* br  = bias_term + (size_t)r * OUT_F;

    // warm the relation-indexed rows while we gather the h row
    __builtin_prefetch(W,   0, 3);
    __builtin_prefetch(gwr, 0, 3);
    __builtin_prefetch(br,  0, 3);

    // gather source row (80B, 16B-aligned -> 5x b128 loads)
    float src[IN_F];
    const float4* hs4 = reinterpret_cast<const float4*>(h + (size_t)s * IN_F);
    #pragma unroll
    for (int q = 0; q < 5; ++q) {
        float4 t = hs4[q];
        src[4*q + 0] = t.x; src[4*q + 1] = t.y;
        src[4*q + 2] = t.z; src[4*q + 3] = t.w;
    }

    // block-diagonal matvec: 4 bases of 5x5 (weight table is L1/L2 hot: 80KB)
    float acc[OUT_F];
    #pragma unroll
    for (int o = 0; o < OUT_F; ++o) acc[o] = 0.0f;
    #pragma unroll
    for (int b = 0; b < 4; ++b) {
        #pragma unroll
        for (int i = 0; i < 5; ++i) {
            float si = src[b*5 + i];
            const float* wr = W + b*25 + i*5;
            #pragma unroll
            for (int o = 0; o < 5; ++o)
                acc[b*5 + o] = fmaf(si, wr[o], acc[b*5 + o]);
        }
    }

    // gate
    float gsum = gate_b[r];
    #pragma unroll
    for (int i = 0; i < IN_F; ++i) gsum = fmaf(src[i], gwr[i], gsum);
    float gate = 1.0f / (1.0f + __expf(-gsum));

    // gated message + bias, scatter-add (global_atomic_add_f32, resolved in L2)
    float* pd = agg + (size_t)d * OUT_F;
    #pragma unroll
    for (int o = 0; o < OUT_F; ++o)
        unsafeAtomicAdd(pd + o, gate * (acc[o] + br[o]));
}

// ---------------------------------------------------------------------------
// Kernel 3: loop_message = h @ loop_weight via V_WMMA_F32_16X16X4_F32,
// fused with out = relu(agg + loop_message). One wave per 16-row tile.
//   A (16x4 f32):  lane = m + 16*(k>=2), vgpr = k&1
//   B (4x16 f32):  vgpr v, lane l -> row = v + 2*(l/16), col = l%16
//   C/D (16x16):   vgpr j, lane l -> row = j + 8*(l/16), col = l%16
// N=20 -> tile0 cols 0..15, tile1 cols 16..19 zero-padded. K=20 -> 5 steps.
// All boundary guards are branchless (clamped address + select) so the 5 A
// loads issue back-to-back and the 10 WMMAs chain without EXEC toggling.
// ---------------------------------------------------------------------------
__global__ __launch_bounds__(256) void loop_relu_wmma_kernel(
    const float* __restrict__ h,    // [n_nodes][20]
    const float* __restrict__ lw,   // [20][20] row-major
    float*       __restrict__ out,  // [n_nodes][20], holds agg on entry
    int n_nodes, int n_tiles)
{
    int wave = (blockIdx.x * blockDim.x + threadIdx.x) >> 5;
    if (wave >= n_tiles) return;          // wave-uniform: EXEC stays full
    int lane = threadIdx.x & 31;
    int m  = lane & 15;
    int hi = lane >> 4;                   // 0 or 1

    // preload B fragments for all 5 K-steps, both N-tiles (branchless col pad)
    bool col1_ok = (16 + m) < OUT_F;      // m < 4
    v2f bf0[5], bf1[5];
    #pragma unroll
    for (int s = 0; s < 5; ++s) {
        #pragma unroll
        for (int v = 0; v < 2; ++v) {
            int kk = 4*s + 2*hi + v;                       // B row (K index)
            bf0[s][v] = lw[kk * OUT_F + m];                // cols 0..15
            float b1 = lw[kk * OUT_F + (col1_ok ? 16 + m : m)];
            bf1[s][v] = col1_ok ? b1 : 0.0f;
        }
    }

    int rbase = wave * 16;
    int arow  = rbase + m;
    bool arow_ok = (arow < n_nodes);
    int safe_row = arow_ok ? arow : (n_nodes - 1);

    // preload all 5 A fragments: unconditional loads, zero via select
    const float* ap = h + (size_t)safe_row * IN_F + 2*hi;  // 8B-aligned
    v2f af[5];
    #pragma unroll
    for (int s = 0; s < 5; ++s) {
        float2 t = *reinterpret_cast<const float2*>(ap + 4*s);
        af[s][0] = arow_ok ? t.x : 0.0f;
        af[s][1] = arow_ok ? t.y : 0.0f;
    }

    v8f c0 = {};
    v8f c1 = {};
    #pragma unroll
    for (int s = 0; s < 5; ++s) {
        c0 = __builtin_amdgcn_wmma_f32_16x16x4_f32(false, af[s], false, bf0[s],
                                                   (short)0, c0, false, false);
        c1 = __builtin_amdgcn_wmma_f32_16x16x4_f32(false, af[s], false, bf1[s],
                                                   (short)0, c1, false, false);
    }

    // epilogue: out = relu(agg + loop_message); each element owned by one lane
    #pragma unroll
    for (int j = 0; j < 8; ++j) {
        int row = rbase + j + 8*hi;
        if (row < n_nodes) {
            float* po = out + (size_t)row * OUT_F;
            float vA = po[m] + c0[j];
            po[m] = vA > 0.0f ? vA : 0.0f;
            if (col1_ok) {
                float vB = po[16 + m] + c1[j];
                po[16 + m] = vB > 0.0f ? vB : 0.0f;
            }
        }
    }
}

// ---------------------------------------------------------------------------
extern "C" void kernel_launch(void* const* d_in, const int* in_sizes, int n_in,
                              void* d_out, int out_size, void* d_ws, size_t ws_size,
                              hipStream_t stream) {
    const float* h         = (const float*)d_in[0];
    const float* weight    = (const float*)d_in[1];
    const float* bias_term = (const float*)d_in[2];
    const float* gate_w    = (const float*)d_in[3];
    const float* gate_b    = (const float*)d_in[4];
    const float* lw        = (const float*)d_in[5];
    const int*   esrc      = (const int*)d_in[6];
    const int*   edst      = (const int*)d_in[7];
    const int*   etype     = (const int*)d_in[8];
    float*       out       = (float*)d_out;

    int n_nodes = in_sizes[0] / IN_F;
    int n_edges = in_sizes[6];
    int n_out   = out_size;          // n_nodes * OUT_F, divisible by 4

    // 1) zero aggregation buffer (d_out doubles as agg scratch)
    int n4 = n_out / 4;
    int zb = (n4 + 1023) / 1024;
    if (zb < 1) zb = 1;
    zero_out_kernel<<<zb, 256, 0, stream>>>((float4*)out, n4);

    // 2) per-edge gather / gate / scatter-add
    edge_kernel<<<(n_edges + 255) / 256, 256, 0, stream>>>(
        h, weight, bias_term, gate_w, gate_b, esrc, edst, etype, out, n_edges);

    // 3) WMMA loop-GEMM + fused ReLU epilogue (one wave per 16-row tile)
    int n_tiles  = (n_nodes + 15) / 16;
    int nthreads = n_tiles * 32;
    loop_relu_wmma_kernel<<<(nthreads + 255) / 256, 256, 0, stream>>>(
        h, lw, out, n_nodes, n_tiles);
}